// LML_4440996184861
// MI455X (gfx1250) — compile-verified
//
#include <hip/hip_runtime.h>

// LML forward for MI455X (gfx1250, wave32).
// Pipeline (all on `stream`, graph-capture safe, deterministic):
//   1) k_minmax   <<<256,256>>> : per-block max/min of x -> ws partials
//   2) k_bracket  <<<1,256>>>   : reduce partials; state = {lo=-max-20, hi=-min+20, r}
//   3) 30x { k_fsum <<<256,256>>> ; k_update <<<1,32>>> }  (branched bisection)
//      k_fsum:   per-block partial sums of sigmoid(x+nu_b), b=0..9 -> P[256][16]
//      k_update: single wave reduces P with V_WMMA_F32_16X16X4_F32 (ones-A),
//                then updates the bracket (ballot/popcount + jnp-style clamp)
//   4) k_final  <<<2048,256>>>  : y = sigmoid(x + (lo + r/2))

#define NX_TOT   2097152
#define NTOPF    1024.0f
#define NBRANCH  10
#define NITER    30
#define RBLOCKS  256
#define RTHREADS 256
#define L2E      1.4426950408889634f

// ws float-offset layout
#define WS_LO    0
#define WS_HI    1
#define WS_R     2
#define WS_PMAX  16                 // 256 floats
#define WS_PMIN  272                // 256 floats
#define WS_P     528                // 256*16 floats (partial sums, cols 10..15 = 0)

typedef __attribute__((ext_vector_type(2))) float v2f;
typedef __attribute__((ext_vector_type(8))) float v8f;

__device__ __constant__ float LSC[NBRANCH] = {
    0.0f, 1.0f/9.0f, 2.0f/9.0f, 3.0f/9.0f, 4.0f/9.0f,
    5.0f/9.0f, 6.0f/9.0f, 7.0f/9.0f, 8.0f/9.0f, 1.0f};

// sigmoid(x+nu) = 1/(1+exp2(zl)), zl = -(x+nu)*log2(e)
__device__ __forceinline__ float sig_from_zl(float zl) {
    float e = __builtin_amdgcn_exp2f(zl);
    return __builtin_amdgcn_rcpf(1.0f + e);
}

// ---------------- 1) per-block max/min ----------------
__global__ void k_minmax(const float* __restrict__ x, float* __restrict__ ws) {
    int tid = blockIdx.x * blockDim.x + threadIdx.x;
    float mx = -3.4e38f, mn = 3.4e38f;
    for (int i = tid; i < NX_TOT; i += RBLOCKS * RTHREADS) {
        float v = x[i];
        mx = fmaxf(mx, v);
        mn = fminf(mn, v);
    }
    for (int off = 16; off; off >>= 1) {
        mx = fmaxf(mx, __shfl_xor(mx, off, 32));
        mn = fminf(mn, __shfl_xor(mn, off, 32));
    }
    __shared__ float smx[8], smn[8];
    int lane = threadIdx.x & 31, wv = threadIdx.x >> 5;
    if (lane == 0) { smx[wv] = mx; smn[wv] = mn; }
    __syncthreads();
    if (threadIdx.x == 0) {
        float bmx = smx[0], bmn = smn[0];
        for (int w = 1; w < 8; ++w) { bmx = fmaxf(bmx, smx[w]); bmn = fminf(bmn, smn[w]); }
        ws[WS_PMAX + blockIdx.x] = bmx;
        ws[WS_PMIN + blockIdx.x] = bmn;
    }
}

// ---------------- 2) bracket init ----------------
__global__ void k_bracket(float* __restrict__ ws) {
    int t = threadIdx.x;
    float mx = ws[WS_PMAX + t];
    float mn = ws[WS_PMIN + t];
    for (int off = 16; off; off >>= 1) {
        mx = fmaxf(mx, __shfl_xor(mx, off, 32));
        mn = fminf(mn, __shfl_xor(mn, off, 32));
    }
    __shared__ float smx[8], smn[8];
    int lane = t & 31, wv = t >> 5;
    if (lane == 0) { smx[wv] = mx; smn[wv] = mn; }
    __syncthreads();
    if (t == 0) {
        float bmx = smx[0], bmn = smn[0];
        for (int w = 1; w < 8; ++w) { bmx = fmaxf(bmx, smx[w]); bmn = fminf(bmn, smn[w]); }
        // generous but guaranteed bracket: sum@lo < 1024 < sum@hi
        float lo = -bmx - 20.0f;
        float hi = -bmn + 20.0f;
        ws[WS_LO] = lo;
        ws[WS_HI] = hi;
        ws[WS_R]  = hi - lo;
    }
}

// ---------------- 3a) branched sigmoid partial sums ----------------
__global__ void k_fsum(const float* __restrict__ x, float* __restrict__ ws) {
    float lo = ws[WS_LO], hi = ws[WS_HI];
    float r  = hi - lo;
    float nul[NBRANCH];
#pragma unroll
    for (int b = 0; b < NBRANCH; ++b) {
        float nu = r * LSC[b] + lo;
        nul[b] = -nu * L2E;                  // folded log2(e) term
    }
    float acc[NBRANCH];
#pragma unroll
    for (int b = 0; b < NBRANCH; ++b) acc[b] = 0.0f;

    int tid = blockIdx.x * RTHREADS + threadIdx.x;
    for (int i = tid * 4; i < NX_TOT; i += RBLOCKS * RTHREADS * 4) {
        float4 v = *(const float4*)(x + i);
        float xs[4] = {v.x, v.y, v.z, v.w};
#pragma unroll
        for (int e = 0; e < 4; ++e) {
            float xl = -L2E * xs[e];
#pragma unroll
            for (int b = 0; b < NBRANCH; ++b)
                acc[b] += sig_from_zl(xl + nul[b]);
        }
    }
    // wave32 butterfly reduce
    for (int off = 16; off; off >>= 1) {
#pragma unroll
        for (int b = 0; b < NBRANCH; ++b)
            acc[b] += __shfl_xor(acc[b], off, 32);
    }
    __shared__ float sred[8][NBRANCH];
    int lane = threadIdx.x & 31, wv = threadIdx.x >> 5;
    if (lane == 0) {
#pragma unroll
        for (int b = 0; b < NBRANCH; ++b) sred[wv][b] = acc[b];
    }
    __syncthreads();
    if (threadIdx.x < 16) {
        float s = 0.0f;
        if (threadIdx.x < NBRANCH)
            for (int w = 0; w < 8; ++w) s += sred[w][threadIdx.x];
        ws[WS_P + blockIdx.x * 16 + threadIdx.x] = s;   // cols 10..15 = 0
    }
}

// ---------------- 3b) WMMA reduce + bracket update (single wave) ----------------
__global__ void k_update(float* __restrict__ ws) {
    int lane = threadIdx.x;           // 0..31, EXEC all ones (required for WMMA)
    v2f a = {1.0f, 1.0f};             // all-ones 16x4 A matrix
    v8f c = {};                       // f32 16x16 accumulator
    const float* __restrict__ P = ws + WS_P;
    int col  = lane & 15;
    int rsel = (lane < 16) ? 0 : 2;
    for (int ch = 0; ch < 64; ++ch) { // 64 * 4 = 256 rows of partials
        int row = ch * 4 + rsel;
        v2f b;
        b.x = P[row * 16 + col];
        b.y = P[(row + 1) * 16 + col];
        // D[m,n] = sum_k B[k,n] + C[m,n] since A==1 -> column sums (n = branch)
        c = __builtin_amdgcn_wmma_f32_16x16x4_f32(
                false, a, false, b, (short)0, c, false, false);
    }
    float fs = ((const float*)&c)[0] - NTOPF;   // row M=0: lanes 0..15 = fs[branch]
    unsigned long long m = __ballot((lane < NBRANCH) && (fs < 0.0f));
    int cnt = __popcll(m);
    if (lane == 0) {
        float lo = ws[WS_LO], hi = ws[WS_HI];
        float r  = hi - lo;
        int il = cnt - 1; if (il < 0) il = 0;           // max(cnt-1, 0)
        int ih = il + 1;  if (ih > NBRANCH - 1) ih = NBRANCH - 1;  // jnp OOB clamp
        ws[WS_LO] = r * LSC[il] + lo;
        ws[WS_HI] = r * LSC[ih] + lo;
        ws[WS_R]  = r;                                   // r from iteration start
    }
}

// ---------------- 4) final elementwise sigmoid ----------------
__global__ void k_final(const float* __restrict__ x, const float* __restrict__ ws,
                        float* __restrict__ y) {
    float nu  = ws[WS_LO] + ws[WS_R] * 0.5f;
    float nul = -nu * L2E;
    int i = (blockIdx.x * 256 + threadIdx.x) * 4;
    if (i + 3 < NX_TOT) {
        float4 v = *(const float4*)(x + i);
        float4 o;
        o.x = sig_from_zl(-L2E * v.x + nul);
        o.y = sig_from_zl(-L2E * v.y + nul);
        o.z = sig_from_zl(-L2E * v.z + nul);
        o.w = sig_from_zl(-L2E * v.w + nul);
        *(float4*)(y + i) = o;
    }
}

extern "C" void kernel_launch(void* const* d_in, const int* in_sizes, int n_in,
                              void* d_out, int out_size, void* d_ws, size_t ws_size,
                              hipStream_t stream) {
    (void)in_sizes; (void)n_in; (void)out_size; (void)ws_size;
    const float* x  = (const float*)d_in[0];
    float*       y  = (float*)d_out;
    float*       ws = (float*)d_ws;

    k_minmax <<<RBLOCKS, RTHREADS, 0, stream>>>(x, ws);
    k_bracket<<<1, 256, 0, stream>>>(ws);
    for (int it = 0; it < NITER; ++it) {
        k_fsum  <<<RBLOCKS, RTHREADS, 0, stream>>>(x, ws);
        k_update<<<1, 32, 0, stream>>>(ws);
    }
    k_final<<<NX_TOT / (256 * 4), 256, 0, stream>>>(x, ws, y);
}